// CGCN_67826123538941
// MI455X (gfx1250) — compile-verified
//
#include <hip/hip_runtime.h>
#include <hip/hip_bf16.h>

#define B_ 64
#define T_ 12
#define K_ 4
#define N_ 400
#define D_ 64
#define E_ 16
#define NK_ (N_ * K_)

typedef float v2f_t __attribute__((ext_vector_type(2)));
typedef float v8f_t __attribute__((ext_vector_type(8)));

#define WMMA_F32_4(A, Bv, C) \
    __builtin_amdgcn_wmma_f32_16x16x4_f32(false, (A), false, (Bv), (short)0, (C), false, false)

// ---------------------------------------------------------------------------
// Kernel A: per-t LN(embeddings) + supports = softmax(embt @ embt^T, axis=1)
// grid = T_, block = 256
// ---------------------------------------------------------------------------
__global__ __launch_bounds__(256) void supports_kernel(
    const float* __restrict__ ne, const float* __restrict__ te,
    const float* __restrict__ g, const float* __restrict__ bb,
    float* __restrict__ sup) {
    const int t = blockIdx.x;
    const int tid = threadIdx.x;
    __shared__ float embt[N_ * 17];  // padded stride 17 (odd) -> conflict-free

    for (int n = tid; n < N_; n += 256) {
        float v[E_];
        float s = 0.f;
        for (int e = 0; e < E_; ++e) { v[e] = ne[n * E_ + e] + te[t * E_ + e]; s += v[e]; }
        float m = s * (1.0f / E_);
        float var = 0.f;
        for (int e = 0; e < E_; ++e) { float d = v[e] - m; var += d * d; }
        var *= (1.0f / E_);
        float r = rsqrtf(var + 1e-12f);
        for (int e = 0; e < E_; ++e) embt[n * 17 + e] = (v[e] - m) * r * g[e] + bb[e];
    }
    __syncthreads();

    const int wave = tid >> 5, lane = tid & 31;
    for (int n = wave; n < N_; n += 8) {
        float a[E_];
        for (int e = 0; e < E_; ++e) a[e] = embt[n * 17 + e];
        float lg[13];
        float mx = -3.4e38f;
        for (int c = 0; c < 13; ++c) {
            int m = c * 32 + lane;
            float d = -3.4e38f;
            if (m < N_) {
                d = 0.f;
                for (int e = 0; e < E_; ++e) d += a[e] * embt[m * 17 + e];
            }
            lg[c] = d;
            mx = fmaxf(mx, d);
        }
        for (int off = 16; off; off >>= 1) mx = fmaxf(mx, __shfl_xor(mx, off, 32));
        float sum = 0.f;
        for (int c = 0; c < 13; ++c) {
            int m = c * 32 + lane;
            float e = (m < N_) ? expf(lg[c] - mx) : 0.f;
            lg[c] = e;
            sum += e;
        }
        for (int off = 16; off; off >>= 1) sum += __shfl_xor(sum, off, 32);
        float inv = 1.0f / sum;
        for (int c = 0; c < 13; ++c) {
            int m = c * 32 + lane;
            if (m < N_) sup[(size_t)t * N_ * N_ + (size_t)n * N_ + m] = lg[c] * inv;
        }
    }
}

// ---------------------------------------------------------------------------
// Kernel B: xg1[b, p=(ic,nc), d] = sum_mc S[nc,mc] * X~[ic,mc,d]
//                                + S[nc,nc] * sum_{jc>=ic} rv[jc]*X~[jc,nc,d]
// with X~[j,m,d] = sx[b, m%4, j*100 + m/4, d]   (reference's scrambled indexing)
// grid = (7 row-tiles of 64, B_), block = 256 (8 waves), per t
// X~ chunk staged K-group-interleaved: xbuf[(K>>2)*1024 + c*4 + (K&3)]
// so each WMMA B-fragment is one contiguous 8B word (conflict-free b64 read).
// ---------------------------------------------------------------------------
__global__ __launch_bounds__(256) void xg1_kernel(
    const float* __restrict__ sx, const float* __restrict__ sup,
    const float* __restrict__ rv, float* __restrict__ xg1, int t) {
    extern __shared__ float smem[];
    float* sbuf = smem;                // 64 x 36 (padded)  S tile
    float* xbuf = sbuf + 64 * 36;      // 8 x 1024          X~ chunk (interleaved)
    float* suf  = xbuf + 8 * 1024;     // 4 x 64 x 64       suffix sums
    float* sdg  = suf + 4 * 64 * 64;   // 64                diag(S)

    const int tid = threadIdx.x;
    const int lane = tid & 31;
    const int wave = tid >> 5;
    const int b = blockIdx.y;
    const int n0 = blockIdx.x * 64;
    const float* supT = sup + (size_t)t * N_ * N_;
    const size_t sxBase = (((size_t)b * T_ + t) * K_) * ((size_t)N_ * D_);

    // ---- load suffix-term sources: suf[jc][ln][d] = X~[jc, n0+ln, d] ----
    for (int idx = tid; idx < 4 * 64 * 64; idx += 256) {
        int jc = idx >> 12;
        int ln = (idx >> 6) & 63;
        int d = idx & 63;
        int n = n0 + ln;
        float v = 0.f;
        if (n < N_) {
            int k = n & 3;
            int nn = jc * 100 + (n >> 2);
            v = sx[sxBase + ((size_t)k * N_ + nn) * D_ + d];
        }
        suf[idx] = v;
    }
    if (tid < 64) {
        int n = n0 + tid;
        sdg[tid] = (n < N_) ? supT[(size_t)n * N_ + n] : 0.f;
    }
    // suffix transform in place (same thread owns (p) for all jc -> no barrier)
    float r0 = rv[t * K_ + 0], r1 = rv[t * K_ + 1];
    float r2 = rv[t * K_ + 2], r3 = rv[t * K_ + 3];
    for (int p = tid; p < 64 * 64; p += 256) {
        float s3 = r3 * suf[3 * 4096 + p];
        float s2 = r2 * suf[2 * 4096 + p] + s3;
        float s1 = r1 * suf[1 * 4096 + p] + s2;
        float s0 = r0 * suf[0 * 4096 + p] + s1;
        suf[0 * 4096 + p] = s0;
        suf[1 * 4096 + p] = s1;
        suf[2 * 4096 + p] = s2;
        suf[3 * 4096 + p] = s3;
    }

    const int rt = wave >> 1;            // row tile 0..3 (16 rows each)
    const int cbase = (wave & 1) * 128;  // column half
    v8f_t acc[8] = {};

    const int arow = (rt * 16 + (lane & 15)) * 36;
    const int koff = (lane >> 4) * 2;    // lanes 0-15: K{0,1}; 16-31: K{2,3}

    for (int mc0 = 0; mc0 < N_; mc0 += 32) {
        __syncthreads();  // protect previous-iteration LDS reads
        // stage S tile: sbuf[ln][j] = S[n0+ln][mc0+j]
        for (int i = 0; i < 8; ++i) {
            int e = i * 256 + tid;
            int ln = e >> 5, j = e & 31;
            int n = n0 + ln;
            if (n >= N_) n = N_ - 1;
            int m = mc0 + j;
            sbuf[ln * 36 + j] = (m < N_) ? supT[(size_t)n * N_ + m] : 0.f;
        }
        // prefetch next S chunk
        if (mc0 + 32 < N_ && tid < 64) {
            int nq = n0 + tid;
            if (nq >= N_) nq = N_ - 1;
            __builtin_prefetch(&supT[(size_t)nq * N_ + mc0 + 32], 0, 1);
        }
        // stage X~ chunk interleaved: 4x4 (m x col) register transpose per task
        for (int it = 0; it < 2; ++it) {
            int task = it * 256 + tid;
            int mg = task >> 6;        // 0..7 K-group within chunk
            int c0 = (task & 63) * 4;  // column block
            int ic = c0 >> 6, d0 = c0 & 63;
            float4 r[4];
            for (int s = 0; s < 4; ++s) {
                int m = mc0 + mg * 4 + s;
                if (m < N_) {
                    int k = m & 3;
                    const float* p =
                        &sx[sxBase + ((size_t)k * N_ + ic * 100 + (m >> 2)) * D_ + d0];
                    r[s] = *(const float4*)p;
                } else {
                    r[s] = make_float4(0.f, 0.f, 0.f, 0.f);
                }
            }
            float* q = &xbuf[mg * 1024 + c0 * 4];
            *(float4*)(q + 0)  = make_float4(r[0].x, r[1].x, r[2].x, r[3].x);
            *(float4*)(q + 4)  = make_float4(r[0].y, r[1].y, r[2].y, r[3].y);
            *(float4*)(q + 8)  = make_float4(r[0].z, r[1].z, r[2].z, r[3].z);
            *(float4*)(q + 12) = make_float4(r[0].w, r[1].w, r[2].w, r[3].w);
        }
        __syncthreads();
        // 8 K-steps of 4, 8 col tiles per wave; fragments are single b64 loads
        for (int kk = 0; kk < 8; ++kk) {
            v2f_t A = *(const v2f_t*)&sbuf[arow + kk * 4 + koff];
            const int bb = kk * 1024 + (lane & 15) * 4 + koff;
            for (int ct = 0; ct < 8; ++ct) {
                v2f_t Bv = *(const v2f_t*)&xbuf[bb + (cbase + ct * 16) * 4];
                acc[ct] = WMMA_F32_4(A, Bv, acc[ct]);
            }
        }
    }

    // store: add diag-suffix term, scatter to xg1[b][p=ic*400+n][d]
    for (int ct = 0; ct < 8; ++ct) {
        int col = cbase + ct * 16 + (lane & 15);
        int ic = col >> 6, d = col & 63;
        for (int v = 0; v < 8; ++v) {
            int ln = rt * 16 + v + ((lane >> 4) << 3);
            int n = n0 + ln;
            if (n < N_) {
                float val = acc[ct][v] + sdg[ln] * suf[ic * 4096 + ln * 64 + d];
                int p = ic * N_ + n;
                xg1[((size_t)b * NK_ + p) * D_ + d] = val;
            }
        }
    }
}

// ---------------------------------------------------------------------------
// Kernel C: per (t, n): for k=0..3 mix W_nk from weights_pool, GEMM
// h = [sx | xg1] (64x128) @ W (128x64) + bias, LN, mean over k,
// then final LN + mean_k(sx), write out[b,t,n,:].
// W staged K-group-interleaved: wbuf[(i>>2)*256 + o*4 + (i&3)].
// grid = N_, block = 256 (8 waves), per t
// ---------------------------------------------------------------------------
__global__ __launch_bounds__(256) void head_kernel(
    const float* __restrict__ sx, const float* __restrict__ xg1,
    const float* __restrict__ ne, const float* __restrict__ te,
    const int* __restrict__ sidx, const float* __restrict__ wp,
    const float* __restrict__ bp, const float* __restrict__ gg,
    const float* __restrict__ gb, const float* __restrict__ og,
    const float* __restrict__ ob, float* __restrict__ out, int t) {
    extern __shared__ float smem[];
    float* wbuf  = smem;                // 32 x 256 mixed weight (interleaved)
    float* xgbuf = wbuf + 32 * 256;     // 64 x 132  [sx | xg1]
    float* hbuf  = xgbuf + 64 * 132;    // 64 x 68   h staging
    float* accb  = hbuf + 64 * 68;      // 64 x 64   LN-mean accumulator
    float* sxab  = accb + 64 * 64;      // 64 x 64   sx sum over k
    float* semb  = sxab + 64 * 64;      // 4 x 16
    float* brow  = semb + 64;           // 64 bias row

    const int tid = threadIdx.x, lane = tid & 31, wave = tid >> 5;
    const int n = blockIdx.x;

    if (tid < 64) {
        int k = tid >> 4, e = tid & 15;
        semb[tid] = ne[n * E_ + e] + te[sidx[t * K_ + k] * E_ + e];
    }
    for (int i = tid; i < 64 * 64; i += 256) { accb[i] = 0.f; sxab[i] = 0.f; }
    __syncthreads();

    const int rt = wave >> 1;           // row tile of b
    const int ct0 = (wave & 1) * 2;     // first col tile
    const int arow_base = (rt * 16 + (lane & 15)) * 132;
    const int koff = (lane >> 4) * 2;

    for (int k = 0; k < K_; ++k) {
        __syncthreads();  // guard prior-iteration LDS reads
        // mix W interleaved: thread computes 4 adjacent i for one o -> float4
        for (int it = 0; it < 8; ++it) {
            int task = it * 256 + tid;
            int ig = task >> 6;         // i-group 0..31
            int o = task & 63;
            int c = ig >> 4;            // (ig*4) >> 6
            int i2b = (ig & 15) * 4;
            float s0 = 0.f, s1 = 0.f, s2 = 0.f, s3 = 0.f;
            for (int e = 0; e < E_; ++e) {
                float se = semb[k * 16 + e];
                const float* wpp = wp + (((size_t)e * 2 + c) * 64 + i2b) * 64 + o;
                s0 += se * wpp[0];
                s1 += se * wpp[64];
                s2 += se * wpp[128];
                s3 += se * wpp[192];
            }
            *(float4*)&wbuf[ig * 256 + o * 4] = make_float4(s0, s1, s2, s3);
        }
        if (tid < 64) {
            float s = 0.f;
            for (int e = 0; e < E_; ++e) s += semb[k * 16 + e] * bp[e * 64 + tid];
            brow[tid] = s;
        }
        // build xgbuf rows (b): cols 0..63 = sx[b,t,k,n,:], 64..127 = xg1[b,n*4+k,:]
        {
            int b = tid >> 2, q = tid & 3;
            if (q < 2) {
                const float* p =
                    &sx[((((size_t)b * T_ + t) * K_ + k) * N_ + n) * D_ + q * 32];
                for (int i = 0; i < 8; ++i) {
                    float4 v = *(const float4*)(p + i * 4);
                    int d0 = q * 32 + i * 4;
                    *(float4*)&xgbuf[b * 132 + d0] = v;
                    sxab[b * 64 + d0 + 0] += v.x;
                    sxab[b * 64 + d0 + 1] += v.y;
                    sxab[b * 64 + d0 + 2] += v.z;
                    sxab[b * 64 + d0 + 3] += v.w;
                }
            } else {
                int pf = n * K_ + k;
                const float* p = &xg1[((size_t)b * NK_ + pf) * D_ + (q - 2) * 32];
                for (int i = 0; i < 8; ++i) {
                    float4 v = *(const float4*)(p + i * 4);
                    *(float4*)&xgbuf[b * 132 + 64 + (q - 2) * 32 + i * 4] = v;
                }
            }
        }
        __syncthreads();
        // GEMM 64(b) x 64(o) x 128(i): 32 K-steps of 4, 2 col tiles per wave
        v8f_t a2[2] = {};
        for (int kk = 0; kk < 32; ++kk) {
            v2f_t A = *(const v2f_t*)&xgbuf[arow_base + kk * 4 + koff];
            const int bb = kk * 256 + (lane & 15) * 4 + koff;
            for (int j = 0; j < 2; ++j) {
                v2f_t Bv = *(const v2f_t*)&wbuf[bb + (ct0 + j) * 64];
                a2[j] = WMMA_F32_4(A, Bv, a2[j]);
            }
        }
        // h tiles + bias -> hbuf
        for (int j = 0; j < 2; ++j) {
            int col = (ct0 + j) * 16 + (lane & 15);
            for (int v = 0; v < 8; ++v) {
                int row = rt * 16 + v + ((lane >> 4) << 3);
                hbuf[row * 68 + col] = a2[j][v] + brow[col];
            }
        }
        __syncthreads();
        // gcn LN per row (eps 1e-5), accumulate 1/K of it
        if (tid < 64) {
            int b = tid;
            float s = 0.f, ss = 0.f;
            for (int o = 0; o < 64; ++o) {
                float h = hbuf[b * 68 + o];
                s += h;
                ss += h * h;
            }
            float m = s * (1.f / 64.f);
            float var = ss * (1.f / 64.f) - m * m;
            float r = rsqrtf(var + 1e-5f);
            for (int o = 0; o < 64; ++o)
                accb[b * 64 + o] +=
                    ((hbuf[b * 68 + o] - m) * r * gg[o] + gb[o]) * 0.25f;
        }
    }
    __syncthreads();
    // final LN (eps 1e-5) + mean_k(sx), write out[b,t,n,:]
    if (tid < 64) {
        int b = tid;
        float s = 0.f, ss = 0.f;
        for (int o = 0; o < 64; ++o) {
            float h = accb[b * 64 + o];
            s += h;
            ss += h * h;
        }
        float m = s * (1.f / 64.f);
        float var = ss * (1.f / 64.f) - m * m;
        float r = rsqrtf(var + 1e-5f);
        for (int o = 0; o < 64; ++o) {
            float v = (accb[b * 64 + o] - m) * r * og[o] + ob[o] +
                      sxab[b * 64 + o] * 0.25f;
            out[(((size_t)b * T_ + t) * N_ + n) * D_ + o] = v;
        }
    }
}

// ---------------------------------------------------------------------------
extern "C" void kernel_launch(void* const* d_in, const int* in_sizes, int n_in,
                              void* d_out, int out_size, void* d_ws, size_t ws_size,
                              hipStream_t stream) {
    (void)in_sizes; (void)n_in; (void)out_size; (void)ws_size;
    // input order: x(unused), selected_x, selected_relevant_values,
    // selected_indices, node_embeddings, time_embeddings, ln_emb_g, ln_emb_b,
    // weights_pool, bias_pool, gcn_ln_g, gcn_ln_b, out_ln_g, out_ln_b
    const float* sx   = (const float*)d_in[1];
    const float* rv   = (const float*)d_in[2];
    const int*   sidx = (const int*)d_in[3];
    const float* ne   = (const float*)d_in[4];
    const float* te   = (const float*)d_in[5];
    const float* lng  = (const float*)d_in[6];
    const float* lnb  = (const float*)d_in[7];
    const float* wp   = (const float*)d_in[8];
    const float* bp   = (const float*)d_in[9];
    const float* gg   = (const float*)d_in[10];
    const float* gb   = (const float*)d_in[11];
    const float* og   = (const float*)d_in[12];
    const float* ob   = (const float*)d_in[13];
    float* out = (float*)d_out;

    float* supports = (float*)d_ws;                       // T*N*N floats (7.7 MB)
    float* xg1 = supports + (size_t)T_ * N_ * N_;         // B*NK*D floats (26.2 MB)

    supports_kernel<<<T_, 256, 0, stream>>>(ne, te, lng, lnb, supports);

    const size_t smemB = (size_t)(64 * 36 + 8 * 1024 + 4 * 64 * 64 + 64) * 4;
    const size_t smemC =
        (size_t)(32 * 256 + 64 * 132 + 64 * 68 + 64 * 64 + 64 * 64 + 64 + 64) * 4;

    for (int t = 0; t < T_; ++t) {
        xg1_kernel<<<dim3(7, B_), 256, smemB, stream>>>(sx, supports, rv, xg1, t);
        head_kernel<<<N_, 256, smemC, stream>>>(sx, xg1, ne, te, sidx, wp, bp,
                                                gg, gb, og, ob, out, t);
    }
}